// ScoreAlign_46583215292962
// MI455X (gfx1250) — compile-verified
//
#include <hip/hip_runtime.h>

// Problem constants from the reference: B=8, n=128, d=512, h=1024
#define BATCH 8
#define NN    128
#define D     512
#define H     1024
#define F2    1024   // 2*d

typedef __attribute__((ext_vector_type(2))) float v2f;
typedef __attribute__((ext_vector_type(8))) float v8f;

// ---------------------------------------------------------------------------
// Kernel 1: fold the two linear layers.
//   w[f] = sum_h W1[f,h] * W2[h]   (f = 0..1023)
//   c    = sum_h b1[h]   * W2[h] + b2
// ws layout: ws[0..1023] = w, ws[1024] = c, ws[1025..2048] = u, ws[2049..3072] = v
// ---------------------------------------------------------------------------
__global__ void fold_w_kernel(const float* __restrict__ W1,
                              const float* __restrict__ b1,
                              const float* __restrict__ W2,
                              const float* __restrict__ b2,
                              float* __restrict__ ws) {
  const int f = blockIdx.x * blockDim.x + threadIdx.x;  // 0..1023
  if (f < F2) {
    const float* row = W1 + (size_t)f * H;
    float s = 0.0f;
    #pragma unroll 4
    for (int hh = 0; hh < H; ++hh) s = fmaf(row[hh], W2[hh], s);
    ws[f] = s;
  }
  if (f == 0) {
    float c = b2[0];
    for (int hh = 0; hh < H; ++hh) c = fmaf(b1[hh], W2[hh], c);
    ws[F2] = c;
  }
}

// ---------------------------------------------------------------------------
// Kernel 2: per-row projections against w2 = ws[512..1023].
//   u[b,i] = a[b,i,:] . w2     v[b,j] = b[b,j,:] . w2
// ---------------------------------------------------------------------------
__global__ void uv_kernel(const float* __restrict__ a,
                          const float* __restrict__ b,
                          float* __restrict__ ws) {
  const int t = blockIdx.x * blockDim.x + threadIdx.x;  // 0..2047
  const float* w2 = ws + D;                 // second half of w
  float* u = ws + F2 + 1;
  float* v = u + BATCH * NN;
  const int idx = t & (BATCH * NN - 1);     // 0..1023
  const float* src = (t < BATCH * NN) ? (a + (size_t)idx * D)
                                      : (b + (size_t)idx * D);
  float s = 0.0f;
  #pragma unroll 4
  for (int k = 0; k < D; ++k) s = fmaf(src[k], w2[k], s);
  if (t < BATCH * NN) u[idx] = s; else v[idx] = s;
}

// ---------------------------------------------------------------------------
// Kernel 3: batched 16x16-tile NT GEMM on the f32 matrix pipe.
//   S[b,i,j] = sum_k (a[b,i,k]*w1[k]) * b[b,j,k] + u[b,i] - v[b,j] + c
// One wave = one 16x16 tile; 128 x V_WMMA_F32_16X16X4_F32 over d=512.
//
// f32 16x4 A fragment layout (ISA 7.12.2): lane L (lo=L&15, hi=L>>4) holds
// row M=lo, VGPR0=K(2*hi), VGPR1=K(2*hi+1) -> one aligned float2 per step.
// B (4x16, K x N) mirrors it with N=lo. C/D: VGPR r -> M=r+8*hi, N=lo.
// ---------------------------------------------------------------------------
__global__ void __launch_bounds__(32)
gemm_wmma_kernel(const float* __restrict__ A,
                 const float* __restrict__ Bm,
                 const float* __restrict__ ws,
                 float* __restrict__ out) {
  const int lane = threadIdx.x;      // 0..31, full wave32
  const int lo   = lane & 15;
  const int hi   = lane >> 4;        // 0 or 1
  const int j0   = blockIdx.x * 16;  // N tile
  const int i0   = blockIdx.y * 16;  // M tile
  const int bb   = blockIdx.z;       // batch

  const float* w1 = ws;              // w[0..511]
  const float  c  = ws[F2];
  const float* u  = ws + F2 + 1;
  const float* v  = u + BATCH * NN;

  const float* arow = A  + ((size_t)bb * NN + (i0 + lo)) * D;
  const float* brow = Bm + ((size_t)bb * NN + (j0 + lo)) * D;

  v8f acc = {};
  #pragma unroll 8
  for (int k = 0; k < D; k += 4) {
    const int kk = k + 2 * hi;                 // this lane's K pair
    v2f wp = *(const v2f*)(w1 + kk);           // broadcast pair of w1
    v2f av = *(const v2f*)(arow + kk);         // A[M=lo][kk, kk+1]
    v2f bv = *(const v2f*)(brow + kk);         // B[K=kk..][N=lo]
    av = av * wp;                              // fold w1 into A (v_pk_mul_f32)
    // 8 args: (neg_a, A, neg_b, B, c_mod, C, reuse_a, reuse_b)
    acc = __builtin_amdgcn_wmma_f32_16x16x4_f32(
        false, av, false, bv, (short)0, acc, false, false);
  }

  // Epilogue: rank-1 correction + bias, then coalesced row stores.
  const float vj = v[bb * NN + j0 + lo];       // N = lo for every acc VGPR
  #pragma unroll
  for (int r = 0; r < 8; ++r) {
    const int m = r + 8 * hi;
    const float val = acc[r] + u[bb * NN + i0 + m] - vj + c;
    out[((size_t)bb * NN + (i0 + m)) * NN + (j0 + lo)] = val;
  }
}

// ---------------------------------------------------------------------------
extern "C" void kernel_launch(void* const* d_in, const int* in_sizes, int n_in,
                              void* d_out, int out_size, void* d_ws, size_t ws_size,
                              hipStream_t stream) {
  const float* a  = (const float*)d_in[0];  // (8,128,512)
  const float* b  = (const float*)d_in[1];  // (8,128,512)
  const float* W1 = (const float*)d_in[2];  // (1024,1024)
  const float* b1 = (const float*)d_in[3];  // (1024,)
  const float* W2 = (const float*)d_in[4];  // (1024,1)
  const float* b2 = (const float*)d_in[5];  // (1,)
  float* ws  = (float*)d_ws;                // ~12.3 KB used
  float* out = (float*)d_out;               // (8,128,128)

  // 1) w = W1 @ W2, c = b1 . W2 + b2        (1024 threads)
  fold_w_kernel<<<dim3(8), dim3(128), 0, stream>>>(W1, b1, W2, b2, ws);
  // 2) u = a @ w2, v = b @ w2               (2048 threads)
  uv_kernel<<<dim3(8), dim3(256), 0, stream>>>(a, b, ws);
  // 3) scores = (a*w1) @ b^T + u - v + c    (512 waves, one 16x16 tile each)
  gemm_wmma_kernel<<<dim3(NN / 16, NN / 16, BATCH), dim3(32), 0, stream>>>(
      a, b, ws, out);
}